// rAIdiologist_44581760532806
// MI455X (gfx1250) — compile-verified
//
#include <hip/hip_runtime.h>
#include <hip/hip_bf16.h>

#define B_SZ   256
#define IN_CH  1280
#define T_LEN  512
#define ITER_LIMIT 100

typedef __attribute__((ext_vector_type(2))) float v2f;
typedef __attribute__((ext_vector_type(8))) float v8f;

// -------------------------------------------------------------------------
// Kernel 1: G[b][t][g] = sum_c x[b][c][t] * W_ih[g][c]   (g padded 12 -> 16)
// Memory-bound: streams 671 MB of x once.  One wave per 64-wide t strip,
// 4 WMMA accumulator tiles sharing a single B fragment per K-chunk.
// A fragment (16x4 f32): lane m (0-15) holds rows M=m, K={0,1}; lanes 16-31
// hold K={2,3}.  B fragment (4x16): lanes 0-15 hold K-rows {0,1}, lanes
// 16-31 K-rows {2,3}, column N = lane&15.
// Weights staged in LDS zero-padded to 16 columns so B fragments feed WMMA
// directly from ds_load with no select.
// -------------------------------------------------------------------------
__global__ __launch_bounds__(128) void gate_gemm_kernel(
    const float* __restrict__ x, const float* __restrict__ W_ih,
    float* __restrict__ G)
{
    // wlds[p*16 + n] = {W_ih[n][2p], W_ih[n][2p+1]}, zero for n >= 12
    __shared__ float2 wlds[(IN_CH / 2) * 16];    // 81,920 bytes

    const int tid = threadIdx.x;
    for (int i = tid; i < (IN_CH / 2) * 16; i += 128) {
        int p = i >> 4;
        int n = i & 15;
        float2 w = make_float2(0.0f, 0.0f);
        if (n < 12) {
            const float* wsrc = W_ih + n * IN_CH + 2 * p;
            w = make_float2(wsrc[0], wsrc[1]);
        }
        wlds[i] = w;
    }
    __syncthreads();

    const int b    = blockIdx.x >> 1;            // 2 blocks per batch elem
    const int tblk = (blockIdx.x & 1) * 256;
    const int wave = tid >> 5;
    const int lane = tid & 31;
    const int m    = lane & 15;
    const int hi   = lane >> 4;                  // 0: K0/K1, 1: K2/K3
    const int tb   = tblk + wave * 64;

    // per-lane A stream base: x[b][c + 2*hi][tb + m], c advances by 4
    const float* xk = x + (size_t)b * IN_CH * T_LEN
                        + (size_t)(hi * 2) * T_LEN + tb + m;

    const float2* wp = wlds + hi * 16 + m;

    v8f a0 = {0,0,0,0,0,0,0,0};
    v8f a1 = a0, a2 = a0, a3 = a0;

    #pragma unroll 8
    for (int c = 0; c < IN_CH; c += 4) {
        float2 w2 = *wp;
        v2f bm; bm.x = w2.x; bm.y = w2.y;

        v2f x0; x0.x = xk[0];  x0.y = xk[T_LEN];
        v2f x1; x1.x = xk[16]; x1.y = xk[16 + T_LEN];
        v2f x2; x2.x = xk[32]; x2.y = xk[32 + T_LEN];
        v2f x3; x3.x = xk[48]; x3.y = xk[48 + T_LEN];

        a0 = __builtin_amdgcn_wmma_f32_16x16x4_f32(false, x0, false, bm,
                                                   (short)0, a0, false, false);
        a1 = __builtin_amdgcn_wmma_f32_16x16x4_f32(false, x1, false, bm,
                                                   (short)0, a1, false, false);
        a2 = __builtin_amdgcn_wmma_f32_16x16x4_f32(false, x2, false, bm,
                                                   (short)0, a2, false, false);
        a3 = __builtin_amdgcn_wmma_f32_16x16x4_f32(false, x3, false, bm,
                                                   (short)0, a3, false, false);

        xk += 4 * T_LEN;
        wp += 32;                                 // (4/2) * 16
    }

    // D layout: VGPR j, lanes 0-15 -> row M=j, lanes 16-31 -> row M=j+8
    float* g0 = G + ((size_t)(b * T_LEN + tb + hi * 8) * 16) + m;
    #pragma unroll
    for (int j = 0; j < 8; ++j) {
        g0[(0 * 16 + j) * 16] = a0[j];
        g0[(1 * 16 + j) * 16] = a1[j];
        g0[(2 * 16 + j) * 16] = a2[j];
        g0[(3 * 16 + j) * 16] = a3[j];
    }
}

// -------------------------------------------------------------------------
// Kernel 2: sequential LSTM scan + adaptive halting, one lane per batch.
// Reads G (L2-resident, 8 MB) via 128-bit loads; weights/biases in regs.
// -------------------------------------------------------------------------
__device__ __forceinline__ float sigm(float v) {
    return 1.0f / (1.0f + __expf(-v));
}

__global__ __launch_bounds__(32) void lstm_scan_kernel(
    const float* __restrict__ G, const float* __restrict__ W_hh,
    const float* __restrict__ b_ih, const float* __restrict__ b_hh,
    float* __restrict__ out)
{
    const int b = blockIdx.x * 32 + threadIdx.x;
    if (b >= B_SZ) return;

    float whh[12][3], bias[12];
    #pragma unroll
    for (int g = 0; g < 12; ++g) {
        bias[g]   = b_ih[g] + b_hh[g];
        whh[g][0] = W_hh[g * 3 + 0];
        whh[g][1] = W_hh[g * 3 + 1];
        whh[g][2] = W_hh[g * 3 + 2];
    }

    const float* Gb = G + (size_t)b * T_LEN * 16;
    float h[3] = {0, 0, 0}, c[3] = {0, 0, 0};

    // 1) full-sequence pass
    for (int t = 0; t < T_LEN; ++t) {
        const float4* gr = (const float4*)(Gb + t * 16);
        float4 q0 = gr[0], q1 = gr[1], q2 = gr[2];
        float gv[12] = {q0.x, q0.y, q0.z, q0.w,
                        q1.x, q1.y, q1.z, q1.w,
                        q2.x, q2.y, q2.z, q2.w};
        float gate[12];
        #pragma unroll
        for (int g = 0; g < 12; ++g)
            gate[g] = gv[g] + bias[g]
                    + whh[g][0] * h[0] + whh[g][1] * h[1] + whh[g][2] * h[2];
        #pragma unroll
        for (int j = 0; j < 3; ++j) {
            float ig = sigm(gate[j]);
            float fg = sigm(gate[3 + j]);
            float gg = tanhf(gate[6 + j]);
            float og = sigm(gate[9 + j]);
            c[j] = fg * c[j] + ig * gg;
            h[j] = og * tanhf(c[j]);
        }
    }

    // 2) adaptive halting loop (fully predicated, ITER_LIMIT steps)
    int   idx      = T_LEN / 2;
    float ofin[3]  = {0, 0, 0};
    bool  active   = true;
    for (int it = 0; it < ITER_LIMIT; ++it) {
        const float4* gr = (const float4*)(Gb + idx * 16);
        float4 q0 = gr[0], q1 = gr[1], q2 = gr[2];
        float gv[12] = {q0.x, q0.y, q0.z, q0.w,
                        q1.x, q1.y, q1.z, q1.w,
                        q2.x, q2.y, q2.z, q2.w};
        float gate[12];
        #pragma unroll
        for (int g = 0; g < 12; ++g)
            gate[g] = gv[g] + bias[g]
                    + whh[g][0] * h[0] + whh[g][1] * h[1] + whh[g][2] * h[2];
        float h2[3], c2[3];
        #pragma unroll
        for (int j = 0; j < 3; ++j) {
            float ig = sigm(gate[j]);
            float fg = sigm(gate[3 + j]);
            float gg = tanhf(gate[6 + j]);
            float og = sigm(gate[9 + j]);
            c2[j] = fg * c[j] + ig * gg;
            h2[j] = og * tanhf(c2[j]);
        }
        float sig1 = sigm(h2[1]);
        float sig2 = sigm(h2[2]);            // sig[-1] with HID=3
        int idx2 = (sig2 > 0.5f) ? (idx + 1) : (idx - 1);
        idx2 &= (T_LEN - 1);                 // mod T (power of two)
        bool stop = sig1 > 0.5f;
        if (active) {
            h[0] = h2[0]; h[1] = h2[1]; h[2] = h2[2];
            c[0] = c2[0]; c[1] = c2[1]; c[2] = c2[2];
            ofin[0] = h2[0]; ofin[1] = h2[1]; ofin[2] = h2[2];
            idx = idx2;
        }
        active = active && !stop;
    }

    out[b * 2 + 0] = ofin[0];
    out[b * 2 + 1] = ofin[1];
}

// -------------------------------------------------------------------------
extern "C" void kernel_launch(void* const* d_in, const int* in_sizes, int n_in,
                              void* d_out, int out_size, void* d_ws, size_t ws_size,
                              hipStream_t stream) {
    const float* x    = (const float*)d_in[0];
    const float* W_ih = (const float*)d_in[1];
    const float* W_hh = (const float*)d_in[2];
    const float* b_ih = (const float*)d_in[3];
    const float* b_hh = (const float*)d_in[4];
    float* out = (float*)d_out;
    float* G   = (float*)d_ws;   // B*T*16 floats = 8 MiB of scratch

    gate_gemm_kernel<<<B_SZ * (T_LEN / 256), 128, 0, stream>>>(x, W_ih, G);
    lstm_scan_kernel<<<(B_SZ + 31) / 32, 32, 0, stream>>>(G, W_hh, b_ih, b_hh, out);
}